// PrototypeContrastiveLoss_90048284328408
// MI455X (gfx1250) — compile-verified
//
#include <hip/hip_runtime.h>
#include <hip/hip_bf16.h>
#include <math.h>

#define C_CLS 7
#define A_DIM 256
#define GMAX  1024

typedef float v2f __attribute__((ext_vector_type(2)));
typedef float v4f __attribute__((ext_vector_type(4)));
typedef float v8f __attribute__((ext_vector_type(8)));

// ---------------------------------------------------------------------------
// Kernel 1: streaming per-class sum. Block = 256 threads = 8 waves.
// Wave w: half = w&1 selects features [half*128, half*128+128), lanes cover
// 4 consecutive floats each (b128 loads). Rows strided by 4 among wave pairs.
// Label is wave-uniform -> readfirstlane + scalar switch: 1 v_add per element.
// ---------------------------------------------------------------------------
__global__ void __launch_bounds__(256) pcl_class_sum(
    const float* __restrict__ feat, const int* __restrict__ labels,
    float* __restrict__ partial, int* __restrict__ partialCnt,
    int N, int rowsPerBlock)
{
  __shared__ float sums[C_CLS * A_DIM];
  __shared__ int   cnts[C_CLS];
  const int t = threadIdx.x;
  for (int i = t; i < C_CLS * A_DIM; i += 256) sums[i] = 0.0f;
  if (t < C_CLS) cnts[t] = 0;
  __syncthreads();

  const int wave = t >> 5;
  const int lane = t & 31;
  const int half = wave & 1;        // which 128-feature half of the row
  const int wr   = wave >> 1;       // 0..3 row phase
  const int col  = half * 128 + lane * 4;

  const int rowStart = blockIdx.x * rowsPerBlock;
  const int rowEnd   = min(N, rowStart + rowsPerBlock);

  v4f a0 = {0,0,0,0}, a1 = a0, a2 = a0, a3 = a0, a4 = a0, a5 = a0, a6 = a0;
  int n0=0,n1=0,n2=0,n3=0,n4=0,n5=0,n6=0;

  for (int r = rowStart + wr; r < rowEnd; r += 4) {
    const int lbl = __builtin_amdgcn_readfirstlane(labels[r]);
    const v4f v = __builtin_nontemporal_load(
        reinterpret_cast<const v4f*>(feat + (size_t)r * A_DIM + col));
    switch (lbl) {            // wave-uniform scalar branch chain
      case 0: a0 += v; n0++; break;
      case 1: a1 += v; n1++; break;
      case 2: a2 += v; n2++; break;
      case 3: a3 += v; n3++; break;
      case 4: a4 += v; n4++; break;
      case 5: a5 += v; n5++; break;
      case 6: a6 += v; n6++; break;
      default: break;         // IGNORE_LABEL / out of range: contributes nothing
    }
  }

#define PCL_MERGE(c, acc, n)                                   \
  {                                                            \
    const int base = (c) * A_DIM + col;                        \
    atomicAdd(&sums[base + 0], (acc).x);                       \
    atomicAdd(&sums[base + 1], (acc).y);                       \
    atomicAdd(&sums[base + 2], (acc).z);                       \
    atomicAdd(&sums[base + 3], (acc).w);                       \
    if (half == 0 && lane == 0) atomicAdd(&cnts[c], (n));      \
  }
  PCL_MERGE(0, a0, n0) PCL_MERGE(1, a1, n1) PCL_MERGE(2, a2, n2)
  PCL_MERGE(3, a3, n3) PCL_MERGE(4, a4, n4) PCL_MERGE(5, a5, n5)
  PCL_MERGE(6, a6, n6)
#undef PCL_MERGE
  __syncthreads();

  float* pb = partial + (size_t)blockIdx.x * (C_CLS * A_DIM);
  for (int i = t; i < C_CLS * A_DIM; i += 256) pb[i] = sums[i];
  if (t < C_CLS) partialCnt[blockIdx.x * C_CLS + t] = cnts[t];
}

// ---------------------------------------------------------------------------
// Kernel 2: reduce per-block partials -> mean (C x A), written to d_out+1.
// One thread per (class, feature) pair; coalesced strided reads.
// ---------------------------------------------------------------------------
__global__ void __launch_bounds__(128) pcl_reduce_mean(
    const float* __restrict__ partial, const int* __restrict__ partialCnt,
    float* __restrict__ outMean, int G)
{
  const int pair = blockIdx.x * 128 + threadIdx.x;
  if (pair >= C_CLS * A_DIM) return;
  const int c = pair / A_DIM;
  float s = 0.0f;
  for (int g = 0; g < G; ++g)
    s += partial[(size_t)g * (C_CLS * A_DIM) + pair];
  int cn = 0;
  for (int g = 0; g < G; ++g)
    cn += partialCnt[g * C_CLS + c];
  const float denom = (cn == 0) ? 1.0f : (float)cn;
  outMean[pair] = s / denom;
}

// ---------------------------------------------------------------------------
// Kernel 3: normalize Proto & mean, logits = mean_n @ Proto_n^T via
// V_WMMA_F32_16X16X4_F32 (wave 0, K-loop of 64), then diag log-softmax loss.
// LDS tiles are padded to 16 rows (rows 7..15 zeroed) so every lane in the
// WMMA wave loads unconditionally: no exec-mask juggling, b64 DS loads.
// A 16x4 f32 layout: lane l<16 holds {A[l][k0],A[l][k0+1]}, lane l>=16 holds
// {A[l-16][k0+2],A[l-16][k0+3]}. B[k][j] = Proto_n[j][k] gives the identical
// per-lane indexing pattern into Proto_n rows.
// ---------------------------------------------------------------------------
__global__ void __launch_bounds__(256) pcl_loss(
    const float* __restrict__ Proto, const float* __restrict__ mean,
    float* __restrict__ outLoss)
{
  __shared__ float Pn[16 * A_DIM];   // rows 7..15 zero padding
  __shared__ float Cn[16 * A_DIM];
  __shared__ float red[256];
  __shared__ float lg[C_CLS * 8];
  __shared__ float perS[8], valS[8];
  const int t = threadIdx.x;

  // zero the padded tiles (16 rows x 256 cols each)
  for (int i = t; i < 16 * A_DIM; i += 256) { Pn[i] = 0.0f; Cn[i] = 0.0f; }
  __syncthreads();

  for (int c = 0; c < C_CLS; ++c) {
    const float pv = Proto[c * A_DIM + t];
    const float mv = mean[c * A_DIM + t];
    red[t] = pv * pv; __syncthreads();
    for (int s = 128; s > 0; s >>= 1) { if (t < s) red[t] += red[t + s]; __syncthreads(); }
    const float np = sqrtf(red[0]); __syncthreads();
    red[t] = mv * mv; __syncthreads();
    for (int s = 128; s > 0; s >>= 1) { if (t < s) red[t] += red[t + s]; __syncthreads(); }
    const float nm = sqrtf(red[0]); __syncthreads();
    Pn[c * A_DIM + t] = pv / fmaxf(np, 1e-12f);
    Cn[c * A_DIM + t] = mv / fmaxf(nm, 1e-12f);
  }
  __syncthreads();

  if (t < 32) {                       // wave 0, EXEC all ones (no divergence inside)
    const int row  = t & 15;
    const int kOff = (t >> 4) * 2;    // kb = k0 + kOff is even -> 8B aligned
    v8f acc = {0.f,0.f,0.f,0.f,0.f,0.f,0.f,0.f};
#pragma unroll 4
    for (int k0 = 0; k0 < A_DIM; k0 += 4) {
      const int kb = k0 + kOff;
      const v2f a = *reinterpret_cast<const v2f*>(&Cn[row * A_DIM + kb]);
      const v2f b = *reinterpret_cast<const v2f*>(&Pn[row * A_DIM + kb]);
      acc = __builtin_amdgcn_wmma_f32_16x16x4_f32(
          false, a, false, b, (short)0, acc, false, false);
    }
    if (t < C_CLS) {                  // lane t holds column t, VGPR v holds row v
      lg[0 * 8 + t] = acc[0]; lg[1 * 8 + t] = acc[1]; lg[2 * 8 + t] = acc[2];
      lg[3 * 8 + t] = acc[3]; lg[4 * 8 + t] = acc[4]; lg[5 * 8 + t] = acc[5];
      lg[6 * 8 + t] = acc[6];
    }
  }
  __syncthreads();

  if (t < C_CLS) {
    float m = -__builtin_inff(), rowsum = 0.0f;
    for (int j = 0; j < C_CLS; ++j) {
      const float x = lg[t * 8 + j];
      m = fmaxf(m, x); rowsum += x;
    }
    float se = 0.0f;
    for (int j = 0; j < C_CLS; ++j) se += expf(lg[t * 8 + j] - m);
    const float lse = m + logf(se);
    const float per = lse - lg[t * 8 + t];
    const float valid = (rowsum != 0.0f) ? 1.0f : 0.0f;
    perS[t] = per * valid; valS[t] = valid;
  }
  __syncthreads();
  if (t == 0) {
    float sp = 0.0f, sv = 0.0f;
    for (int c = 0; c < C_CLS; ++c) { sp += perS[c]; sv += valS[c]; }
    outLoss[0] = sp / fmaxf(sv, 1.0f);
  }
}

// ---------------------------------------------------------------------------
extern "C" void kernel_launch(void* const* d_in, const int* in_sizes, int n_in,
                              void* d_out, int out_size, void* d_ws, size_t ws_size,
                              hipStream_t stream) {
  const float* Proto  = (const float*)d_in[0];
  const float* feat   = (const float*)d_in[1];
  const int*   labels = (const int*)d_in[2];
  float* out = (float*)d_out;        // [0]=loss, [1..1792]=mean

  const int N = in_sizes[2];

  const size_t bytesPerBlock = (size_t)(C_CLS * A_DIM) * sizeof(float)
                             + (size_t)C_CLS * sizeof(int);
  int G = (int)(ws_size / bytesPerBlock);
  if (G > GMAX) G = GMAX;
  if (G < 1) G = 1;

  float* partial    = (float*)d_ws;
  int*   partialCnt = (int*)((char*)d_ws + (size_t)G * (C_CLS * A_DIM) * sizeof(float));
  const int rowsPerBlock = (N + G - 1) / G;

  pcl_class_sum<<<G, 256, 0, stream>>>(feat, labels, partial, partialCnt,
                                       N, rowsPerBlock);
  pcl_reduce_mean<<<(C_CLS * A_DIM + 127) / 128, 128, 0, stream>>>(
      partial, partialCnt, out + 1, G);
  pcl_loss<<<1, 256, 0, stream>>>(Proto, out + 1, out);
}